// RankingLoss_89962384982542
// MI455X (gfx1250) — compile-verified
//
#include <hip/hip_runtime.h>

typedef __attribute__((ext_vector_type(2))) float v2f;
typedef __attribute__((ext_vector_type(8))) float v8f;

#define RN 8192
#define RMARGIN 1.0f
#define RTILE 16
#define RJSPLIT 8
#define RWAVES_PER_BLOCK 8
#define RNBLOCKS (((RN / RTILE) * RJSPLIT) / RWAVES_PER_BLOCK)  // 512
#define RJCHUNK (RN / RJSPLIT)                                   // 1024

// Each wave owns a 16-row i-tile and a 1024-wide j strip.
// One V_WMMA_F32_16X16X4_F32 per 16x16 tile:
//   A = [p_i, 1, 0, 0] (per row), B = [-1, p_j, x, x] (per col), C = MARGIN
//   D[m][n] = MARGIN - (p_i[m] - p_j[n])
// (A's K=2,3 rows are zero, so B's hi-half lanes are don't-cares.)
__global__ __launch_bounds__(256) void RankingLoss_main(
    const float* __restrict__ preds,
    const float* __restrict__ targets,   // (N,2) interleaved: dur, event
    float* __restrict__ lossPart,
    unsigned* __restrict__ cntPart) {
  const int tid = threadIdx.x;
  const int lane = tid & 31;
  const int waveInBlk = tid >> 5;
  const int waveId = blockIdx.x * RWAVES_PER_BLOCK + waveInBlk;
  const int iTile = waveId >> 3;            // 0..511
  const int jChunk = waveId & (RJSPLIT - 1);
  const int i0 = iTile * RTILE;
  const int jBase = jChunk * RJCHUNK;

  const int lanelo = lane & 15;
  const bool hiHalf = lane >= 16;
  const float INFF = __uint_as_float(0x7f800000u);

  // A matrix (16x4 f32): lanes 0-15 hold K=0 (a.x), K=1 (a.y);
  // lanes 16-31 hold K=2,K=3 which must be zero.
  v2f a;
  a.x = hiHalf ? 0.0f : preds[i0 + lanelo];  // A[m][0] = p_i[m]
  a.y = hiHalf ? 0.0f : 1.0f;                // A[m][1] = 1

  // C input = MARGIN broadcast (folds into WMMA; becomes inline constant).
  v8f cM;
#pragma unroll
  for (int r = 0; r < 8; ++r) cM[r] = RMARGIN;

  // Event folded into duration: if event_i != 1, row i never forms a pair.
  float durIp[8];
#pragma unroll
  for (int r = 0; r < 8; ++r) {
    int m = i0 + r + (hiHalf ? 8 : 0);       // this lane's D row for VGPR r
    float d = targets[2 * m];
    float e = targets[2 * m + 1];
    durIp[r] = (e == 1.0f) ? d : INFF;       // mask == (durIp < durJ)
  }

  // 4 accumulators to break the serial add chain.
  float acc0 = 0.0f, acc1 = 0.0f, acc2 = 0.0f, acc3 = 0.0f;
  unsigned cntAcc = 0;                       // wave-uniform (SALU popcount)

#pragma unroll 4
  for (int j0 = jBase; j0 < jBase + RJCHUNK; j0 += RTILE) {
    // Column data for N = lanelo (L1-resident broadcast loads).
    float pJ   = preds[j0 + lanelo];
    float durJ = targets[2 * (j0 + lanelo)];

    // B (4x16): K=0 row = -1, K=1 row = p_j. Hi-half lanes are don't-cares.
    v2f b;
    b.x = -1.0f;
    b.y = pJ;

    v8f d = __builtin_amdgcn_wmma_f32_16x16x4_f32(
        /*neg_a=*/false, a, /*neg_b=*/false, b,
        /*c_mod=*/(short)0, cM, /*reuse_a=*/false, /*reuse_b=*/false);

#pragma unroll
    for (int r = 0; r < 8; ++r) {
      bool m = durIp[r] < durJ;                        // 1 v_cmp -> SGPR
      float t = m ? d[r] : 0.0f;                       // 1 v_cndmask
      // ReLU as a single v_med3_num_f32 (no canonicalize double-max):
      float pen = __builtin_amdgcn_fmed3f(t, 0.0f, INFF);
      if ((r & 3) == 0)      acc0 += pen;              // 1 v_add_f32
      else if ((r & 3) == 1) acc1 += pen;
      else if ((r & 3) == 2) acc2 += pen;
      else                   acc3 += pen;
      cntAcc += __builtin_popcount(
          __builtin_amdgcn_ballot_w32(m));             // SALU: s_bcnt + s_add
    }
  }

  float lossAcc = (acc0 + acc1) + (acc2 + acc3);

  // Loss: wave32 shuffle reduce (deterministic). Count is already uniform.
#pragma unroll
  for (int off = 16; off >= 1; off >>= 1)
    lossAcc += __shfl_down(lossAcc, off, 32);

  __shared__ float    sLoss[RWAVES_PER_BLOCK];
  __shared__ unsigned sCnt[RWAVES_PER_BLOCK];
  if (lane == 0) { sLoss[waveInBlk] = lossAcc; sCnt[waveInBlk] = cntAcc; }
  __syncthreads();
  if (tid == 0) {
    float L = 0.0f; unsigned C = 0;
    for (int w = 0; w < RWAVES_PER_BLOCK; ++w) { L += sLoss[w]; C += sCnt[w]; }
    lossPart[blockIdx.x] = L;   // every slot written every launch: no ws init
    cntPart[blockIdx.x]  = C;
  }
}

// Finalize: single wave, fixed-order strided sum over 512 partials -> scalar.
__global__ void RankingLoss_final(const float* __restrict__ lossPart,
                                  const unsigned* __restrict__ cntPart,
                                  float* __restrict__ out) {
  int lane = threadIdx.x;  // 32 threads
  float L = 0.0f;
  unsigned long long C = 0;
  for (int i = lane; i < RNBLOCKS; i += 32) {
    L += lossPart[i];
    C += (unsigned long long)cntPart[i];
  }
#pragma unroll
  for (int off = 16; off >= 1; off >>= 1) {
    L += __shfl_down(L, off, 32);
    C += __shfl_down(C, off, 32);
  }
  if (lane == 0) out[0] = (C > 0) ? (L / (float)C) : 0.0f;
}

extern "C" void kernel_launch(void* const* d_in, const int* in_sizes, int n_in,
                              void* d_out, int out_size, void* d_ws, size_t ws_size,
                              hipStream_t stream) {
  (void)in_sizes; (void)n_in; (void)out_size; (void)ws_size;
  const float* preds   = (const float*)d_in[0];   // 8192 f32
  const float* targets = (const float*)d_in[1];   // 8192x2 f32 (dur, event)
  float* lossPart = (float*)d_ws;
  unsigned* cntPart = (unsigned*)((char*)d_ws + RNBLOCKS * sizeof(float));

  RankingLoss_main<<<RNBLOCKS, 256, 0, stream>>>(preds, targets, lossPart, cntPart);
  RankingLoss_final<<<1, 32, 0, stream>>>(lossPart, cntPart, (float*)d_out);
}